// Efron_loss_89833535963286
// MI455X (gfx1250) — compile-verified
//
#include <hip/hip_runtime.h>
#include <math.h>

#define TMAX 1024
#define D_CAP 12288u
#define ACC_BLOCKS 1024
#define THREADS 256

typedef __attribute__((ext_vector_type(2))) float v2f;
typedef __attribute__((ext_vector_type(8))) float v8f;

// ---------------- workspace zeroing (graph-replay safe) ----------------
__global__ void __launch_bounds__(THREADS) efron_zero(float* wsS, float* wsR,
                                                      float* wsTR,
                                                      unsigned int* wsD) {
  int i = blockIdx.x * blockDim.x + threadIdx.x;
  if (i < TMAX) {
    wsS[i] = 0.0f;
    wsR[i] = 0.0f;
    wsTR[i] = 0.0f;
    wsD[i] = 0u;
  }
}

// ---------------- phase 1: segmented sums via LDS privatization --------
__global__ void __launch_bounds__(THREADS) efron_accum(
    const int* __restrict__ times, const int* __restrict__ events,
    const float* __restrict__ risk, float* __restrict__ wsS,
    float* __restrict__ wsR, float* __restrict__ wsTR,
    unsigned int* __restrict__ wsD, int n) {
  __shared__ float sS[TMAX];
  __shared__ float sR[TMAX];
  __shared__ float sTR[TMAX];
  __shared__ unsigned int sD[TMAX];
  for (int i = threadIdx.x; i < TMAX; i += THREADS) {
    sS[i] = 0.0f;
    sR[i] = 0.0f;
    sTR[i] = 0.0f;
    sD[i] = 0u;
  }
  __syncthreads();

  const int gid = blockIdx.x * blockDim.x + threadIdx.x;
  const int stride = gridDim.x * blockDim.x;
  const int n4 = n >> 2;
  const int4* __restrict__ t4 = (const int4*)times;
  const int4* __restrict__ e4 = (const int4*)events;
  const float4* __restrict__ r4 = (const float4*)risk;

  for (int i = gid; i < n4; i += stride) {
    // prefetch one grid-stride ahead (global_prefetch_b8 on gfx1250)
    if (i + stride < n4) {
      __builtin_prefetch((const void*)&t4[i + stride], 0, 3);
      __builtin_prefetch((const void*)&e4[i + stride], 0, 3);
      __builtin_prefetch((const void*)&r4[i + stride], 0, 3);
    }
    int4 tt = t4[i];
    int4 ee = e4[i];
    float4 rr = r4[i];
#define PROC(TT, EE, RR)                          \
  {                                               \
    int t = (TT) & (TMAX - 1);                    \
    float r = (RR);                               \
    float e = expf(r);                            \
    atomicAdd(&sS[t], e);                         \
    if ((EE) == 1) {                              \
      atomicAdd(&sR[t], e);                       \
      atomicAdd(&sTR[t], r);                      \
      atomicAdd(&sD[t], 1u);                      \
    }                                             \
  }
    PROC(tt.x, ee.x, rr.x)
    PROC(tt.y, ee.y, rr.y)
    PROC(tt.z, ee.z, rr.z)
    PROC(tt.w, ee.w, rr.w)
  }
  // scalar tail (n not divisible by 4)
  for (int i = (n4 << 2) + gid; i < n; i += stride) {
    PROC(times[i], events[i], risk[i])
  }
#undef PROC
  __syncthreads();
  for (int i = threadIdx.x; i < TMAX; i += THREADS) {
    atomicAdd(&wsS[i], sS[i]);
    atomicAdd(&wsR[i], sR[i]);
    atomicAdd(&wsTR[i], sTR[i]);
    atomicAdd(&wsD[i], sD[i]);
  }
}

// ---------------- phase 2: per-bin Efron log terms + WMMA reduction ----
__global__ void __launch_bounds__(THREADS) efron_loss(
    const float* __restrict__ wsS, const float* __restrict__ wsR,
    const float* __restrict__ wsTR, const unsigned int* __restrict__ wsD,
    float* __restrict__ wsBin, float* __restrict__ out) {
  const int t = blockIdx.x;
  const float S = wsS[t];
  const float R = wsR[t];
  const float TR = wsTR[t];
  const unsigned int d = wsD[t];
  const unsigned int dcap = d < D_CAP ? d : D_CAP;  // reference masks j < min(d, D_MAX)
  const float inv_d = 1.0f / fmaxf((float)d, 1.0f);

  float local = 0.0f;
  for (unsigned int j = threadIdx.x; j < dcap; j += THREADS) {
    float arg = S - ((float)j * inv_d) * R;
    local += logf(arg);
  }

  __shared__ float part[THREADS];
  part[threadIdx.x] = local;
  __syncthreads();

  float total;
#if __has_builtin(__builtin_amdgcn_wmma_f32_16x16x4_f32)
  // Reduce 256 partials with 4 chained V_WMMA_F32_16X16X4_F32 (B = ones):
  // A[m][k] = part[64c + 4m + k]; D[m][n] = sum_k A[m][k] + C  (n-invariant).
  // All 8 waves run it redundantly so EXEC is all-1s for every WMMA issue.
  {
    const int lane = threadIdx.x & 31;
    const int base = (lane < 16) ? (4 * lane) : (4 * (lane - 16) + 2);
    v8f acc = {0.f, 0.f, 0.f, 0.f, 0.f, 0.f, 0.f, 0.f};
    v2f ones;
    ones.x = 1.0f;
    ones.y = 1.0f;
#pragma unroll
    for (int c = 0; c < 4; ++c) {
      v2f a;
      a.x = part[c * 64 + base];
      a.y = part[c * 64 + base + 1];
      acc = __builtin_amdgcn_wmma_f32_16x16x4_f32(false, a, false, ones,
                                                  (short)0, acc, false, false);
    }
    // lane0 holds sum of rows 0..7 at N=0; lane16 holds rows 8..15 at N=0
    float s = acc[0] + acc[1] + acc[2] + acc[3] + acc[4] + acc[5] + acc[6] +
              acc[7];
    s += __shfl_xor(s, 16, 32);
    total = s;
  }
#else
  // fallback: LDS tree
  __syncthreads();
  for (int off = THREADS / 2; off > 0; off >>= 1) {
    if ((int)threadIdx.x < off) part[threadIdx.x] += part[threadIdx.x + off];
    __syncthreads();
  }
  total = part[0];
#endif

  if (threadIdx.x == 0) {
    wsBin[t] = total - TR;          // per-bin: sum(log terms) - tie_risk_sum
    out[1 + t] = (float)d;          // tie_count_list
    out[1 + TMAX + t] = S;          // cum_exp_risk
    out[1 + 2 * TMAX + t] = (float)t;  // failure_time
  }
}

// ---------------- phase 3: deterministic final reduction --------------
__global__ void __launch_bounds__(THREADS) efron_final(
    const float* __restrict__ wsBin, float* __restrict__ out) {
  __shared__ float sm[THREADS];
  float s = 0.0f;
  for (int i = threadIdx.x; i < TMAX; i += THREADS) s += wsBin[i];
  sm[threadIdx.x] = s;
  __syncthreads();
  for (int off = THREADS / 2; off > 0; off >>= 1) {
    if ((int)threadIdx.x < off) sm[threadIdx.x] += sm[threadIdx.x + off];
    __syncthreads();
  }
  if (threadIdx.x == 0) out[0] = sm[0];
}

extern "C" void kernel_launch(void* const* d_in, const int* in_sizes, int n_in,
                              void* d_out, int out_size, void* d_ws,
                              size_t ws_size, hipStream_t stream) {
  const int* times = (const int*)d_in[0];
  const int* events = (const int*)d_in[1];
  const float* risk = (const float*)d_in[2];
  float* out = (float*)d_out;
  const int n = in_sizes[0];

  // workspace layout: S[1024] R[1024] TR[1024] | D[1024] (u32) | binloss[1024]
  float* wsS = (float*)d_ws;
  float* wsR = wsS + TMAX;
  float* wsTR = wsR + TMAX;
  unsigned int* wsD = (unsigned int*)(wsTR + TMAX);
  float* wsBin = (float*)(wsD + TMAX);

  efron_zero<<<(TMAX + THREADS - 1) / THREADS, THREADS, 0, stream>>>(
      wsS, wsR, wsTR, wsD);
  efron_accum<<<ACC_BLOCKS, THREADS, 0, stream>>>(times, events, risk, wsS,
                                                  wsR, wsTR, wsD, n);
  efron_loss<<<TMAX, THREADS, 0, stream>>>(wsS, wsR, wsTR, wsD, wsBin, out);
  efron_final<<<1, THREADS, 0, stream>>>(wsBin, out);
}